// PPGNLayer_73916387164334
// MI455X (gfx1250) — compile-verified
//
#include <hip/hip_runtime.h>
#include <hip/hip_bf16.h>

// PPGN layer, CDNA5 (gfx1250), wave32.
// Pipeline:
//   K0: map[N*N] = -1
//   K1: map[i*N+k] = e  (edge-id scatter; edges unique per reference)
//   K2: X = C @ W_L1^T, Y = C @ W_L2^T          (f32 WMMA 16x16x4)
//   K3: TM[e,f] = sum_j resx[f,i_e,j]*resy[f,j,k_e]  via sparse 2-hop
//       path matching (ballot scan over j, ~2 matches/edge)
//   K4: out = relu([C|TM] @ W_mlp1^T) @ W_mlp2^T (f32 WMMA, fused via LDS)

#define NN   512
#define EE   16384
#define FF   64
#define FFO  64

typedef __attribute__((ext_vector_type(2))) float v2f;
typedef __attribute__((ext_vector_type(8))) float v8f;

__device__ __forceinline__ v8f wmma4(v2f a, v2f b, v8f c) {
  // D = A(16x4 f32) * B(4x16 f32) + C(16x16 f32)
  return __builtin_amdgcn_wmma_f32_16x16x4_f32(
      /*neg_a=*/false, a, /*neg_b=*/false, b,
      /*c_mod=*/(short)0, c, /*reuse_a=*/false, /*reuse_b=*/false);
}

// ---------------- K0: init edge-id map ----------------
__global__ void k_init_map(int* __restrict__ map) {
  int idx = blockIdx.x * blockDim.x + threadIdx.x;
  if (idx < NN * NN) map[idx] = -1;
}

// ---------------- K1: scatter edge ids ----------------
__global__ void k_scatter_map(const int* __restrict__ ei, int* __restrict__ map) {
  int e = blockIdx.x * blockDim.x + threadIdx.x;
  if (e < EE) {
    int i = ei[e];        // edge_index[0][e]
    int k = ei[EE + e];   // edge_index[1][e]
    map[i * NN + k] = e;
  }
}

// ---------------- K2: X = C@W1^T, Y = C@W2^T (WMMA f32) ----------------
// grid.x = EE/16, block = 64 (2 waves; wave0 -> X/W1, wave1 -> Y/W2)
__global__ void k_gemm_xy(const float* __restrict__ C,
                          const float* __restrict__ W1,
                          const float* __restrict__ W2,
                          float* __restrict__ X, float* __restrict__ Y) {
  const int lane = threadIdx.x & 31;
  const int wv   = threadIdx.x >> 5;
  const int e0   = blockIdx.x * 16;
  const int m    = lane & 15;
  const int kp   = lane >> 4;          // K half (0: K=0..1, 1: K=2..3)
  const float* __restrict__ W = wv ? W2 : W1;
  float* __restrict__ O       = wv ? Y : X;

  v8f acc0 = {}, acc1 = {}, acc2 = {}, acc3 = {};
#pragma unroll
  for (int ks = 0; ks < 16; ++ks) {
    const int k0 = ks * 4 + kp * 2;
    v2f a  = *(const v2f*)(C + (e0 + m) * FF + k0);
    v2f b0 = *(const v2f*)(W + (0 * 16 + m) * FF + k0);  // B(k,n)=W[n][k]
    v2f b1 = *(const v2f*)(W + (1 * 16 + m) * FF + k0);
    v2f b2 = *(const v2f*)(W + (2 * 16 + m) * FF + k0);
    v2f b3 = *(const v2f*)(W + (3 * 16 + m) * FF + k0);
    acc0 = wmma4(a, b0, acc0);
    acc1 = wmma4(a, b1, acc1);
    acc2 = wmma4(a, b2, acc2);
    acc3 = wmma4(a, b3, acc3);
  }
#pragma unroll
  for (int r = 0; r < 8; ++r) {
    const int row = (e0 + r + 8 * kp) * FF;
    O[row + 0 * 16 + m] = acc0[r];
    O[row + 1 * 16 + m] = acc1[r];
    O[row + 2 * 16 + m] = acc2[r];
    O[row + 3 * 16 + m] = acc3[r];
  }
}

// ---------------- K3: sparse 2-hop contraction ----------------
// one wave per edge; lanes scan j in chunks of 32, ballot the matches,
// then all 32 lanes accumulate 2 channels each (f = 2*lane, 2*lane+1).
__global__ void k_pair_contract(const int* __restrict__ ei,
                                const float* __restrict__ X,
                                const float* __restrict__ Y,
                                const int* __restrict__ map,
                                float* __restrict__ TM) {
  const int wave = (blockIdx.x * blockDim.x + threadIdx.x) >> 5;
  const int lane = threadIdx.x & 31;
  if (wave >= EE) return;
  const int i = ei[wave];
  const int k = ei[EE + wave];

  float ax = 0.f, ay = 0.f;
#pragma unroll 4
  for (int jb = 0; jb < NN; jb += 32) {
    const int j  = jb + lane;
    const int e1 = map[i * NN + j];
    const int e2 = map[j * NN + k];
    unsigned long long mask = __ballot((e1 >= 0) && (e2 >= 0));
    while (mask) {
      const int s = __ffsll((unsigned long long)mask) - 1;
      mask &= mask - 1;
      const int a = __shfl(e1, s, 32);
      const int b = __shfl(e2, s, 32);
      v2f xv = *(const v2f*)(X + a * FF + 2 * lane);
      v2f yv = *(const v2f*)(Y + b * FF + 2 * lane);
      ax += xv.x * yv.x;
      ay += xv.y * yv.y;
    }
  }
  v2f r; r.x = ax; r.y = ay;
  *(v2f*)(TM + wave * FF + 2 * lane) = r;
}

// ---------------- K4: fused MLP (WMMA f32, hidden tile in LDS) ----------------
// grid.x = EE/16, block = 32 (one wave, 16 edges). H tile 16x512 stays in LDS.
__global__ void k_mlp(const float* __restrict__ C,
                      const float* __restrict__ TM,
                      const float* __restrict__ Wm1,   // [512,128]
                      const float* __restrict__ Wm2,   // [64,512]
                      float* __restrict__ out) {
  __shared__ float Hs[16][516];                        // stride 516: conflict-spread
  const int lane = threadIdx.x & 31;
  const int e0   = blockIdx.x * 16;
  const int m    = lane & 15;
  const int kp   = lane >> 4;

  // Preload A rows of tmp = [C | TM] (K=128 -> 32 k-steps of 4).
  v2f areg[32];
#pragma unroll
  for (int ks = 0; ks < 32; ++ks) {
    const int k0 = ks * 4 + kp * 2;
    areg[ks] = (ks < 16) ? *(const v2f*)(C + (e0 + m) * FF + k0)
                         : *(const v2f*)(TM + (e0 + m) * FF + (k0 - 64));
  }

  // GEMM1 + relu -> LDS : H[16 x 512]
  for (int t = 0; t < 32; ++t) {
    v8f acc = {};
    const float* __restrict__ wr = Wm1 + (t * 16 + m) * 128;  // B(k,n)=Wm1[n][k]
#pragma unroll
    for (int ks = 0; ks < 32; ++ks) {
      const int k0 = ks * 4 + kp * 2;
      v2f b = *(const v2f*)(wr + k0);
      acc = wmma4(areg[ks], b, acc);
    }
#pragma unroll
    for (int r = 0; r < 8; ++r) {
      const float v = acc[r];
      Hs[r + 8 * kp][t * 16 + m] = v > 0.f ? v : 0.f;
    }
  }
  __syncthreads();

  // GEMM2: out tile 16x64, K=512 from LDS.
#pragma unroll
  for (int t2 = 0; t2 < 4; ++t2) {
    v8f acc = {};
    const float* __restrict__ wr = Wm2 + (t2 * 16 + m) * 512;
#pragma unroll 8
    for (int ks = 0; ks < 128; ++ks) {
      const int k0 = ks * 4 + kp * 2;
      v2f a = *(const v2f*)(&Hs[m][k0]);                // ds_load_b64
      v2f b = *(const v2f*)(wr + k0);
      acc = wmma4(a, b, acc);
    }
#pragma unroll
    for (int r = 0; r < 8; ++r)
      out[(e0 + r + 8 * kp) * FFO + t2 * 16 + m] = acc[r];
  }
}

extern "C" void kernel_launch(void* const* d_in, const int* in_sizes, int n_in,
                              void* d_out, int out_size, void* d_ws, size_t ws_size,
                              hipStream_t stream) {
  const int*   ei  = (const int*)d_in[0];    // [2, E]
  const float* C   = (const float*)d_in[1];  // [E, F]
  // d_in[2] = batch_node (only carries N; unused numerically)
  const float* W1  = (const float*)d_in[3];  // [F, F]
  const float* W2  = (const float*)d_in[4];  // [F, F]
  const float* Wm1 = (const float*)d_in[5];  // [8F, 2F] = [512,128]
  const float* Wm2 = (const float*)d_in[6];  // [FO, 8F] = [64,512]
  float* out = (float*)d_out;

  float* Xb = (float*)d_ws;                  // [E, F]
  float* Yb = Xb + (size_t)EE * FF;          // [E, F]
  float* TM = Yb + (size_t)EE * FF;          // [E, F]
  int*   map = (int*)(TM + (size_t)EE * FF); // [N, N]

  k_init_map<<<(NN * NN) / 256, 256, 0, stream>>>(map);
  k_scatter_map<<<EE / 256, 256, 0, stream>>>(ei, map);
  k_gemm_xy<<<EE / 16, 64, 0, stream>>>(C, W1, W2, Xb, Yb);
  k_pair_contract<<<(EE * 32) / 256, 256, 0, stream>>>(ei, Xb, Yb, map, TM);
  k_mlp<<<EE / 16, 32, 0, stream>>>(C, TM, Wm1, Wm2, out);
}